// UnweightedEncoder_53781580480952
// MI455X (gfx1250) — compile-verified
//
#include <hip/hip_runtime.h>

#define IN_CH 256
#define HID   64

typedef __attribute__((ext_vector_type(2))) float v2f;
typedef __attribute__((ext_vector_type(8))) float v8f;

// ---------------------------------------------------------------------------
// 1) zero the accumulator (d_out) each launch
// ---------------------------------------------------------------------------
__global__ void __launch_bounds__(256) zero_f4(float4* __restrict__ p, int n4) {
  int i = blockIdx.x * blockDim.x + threadIdx.x;
  if (i < n4) p[i] = make_float4(0.f, 0.f, 0.f, 0.f);
}

// ---------------------------------------------------------------------------
// 2) xw = x @ W using V_WMMA_F32_16X16X4_F32 (fp32, full precision; the GEMM
//    is HBM-bandwidth bound so there is no benefit to lower precision).
//    One wave owns a 16(M) x 64(N) tile = 4 x v8f accumulators.
//    W is staged in LDS pair-swizzled so a lane's B fragment for one K-step
//    is a single float2 (ds_load_b64):
//      sW[p*HID + n] = { W[2p][n], W[2p+1][n] }
//    Fragment layouts per CDNA5 ISA:
//      A 16x4 f32 : lanes 0-15 -> (K0,K1), lanes 16-31 -> (K2,K3), M = lane&15
//      B 4x16 f32 : lanes 0-15 -> (K0,K1), lanes 16-31 -> (K2,K3), N = lane&15
//      C 16x16 f32: VGPR v, lanes 0-15 -> M=v, lanes 16-31 -> M=v+8, N=lane&15
// ---------------------------------------------------------------------------
__global__ void __launch_bounds__(256) gemm_xw_wmma(const float* __restrict__ x,
                                                    const float* __restrict__ W,
                                                    float* __restrict__ xw,
                                                    int nNodes) {
  __shared__ float2 sW[(IN_CH / 2) * HID];  // 64 KB

  for (int p = threadIdx.x; p < (IN_CH / 2) * HID; p += blockDim.x) {
    int pk = p / HID;     // K-pair index 0..127
    int n  = p % HID;     // output channel
    float2 w2;
    w2.x = W[(2 * pk + 0) * HID + n];
    w2.y = W[(2 * pk + 1) * HID + n];
    sW[p] = w2;
  }
  __syncthreads();

  const int wave  = threadIdx.x >> 5;
  const int lane  = threadIdx.x & 31;
  const int l15   = lane & 15;
  const int hi    = lane >> 4;                    // 0: K0/K1 half, 1: K2/K3 half
  const int mBase = (blockIdx.x * 8 + wave) * 16; // wave's 16-row tile
  if (mBase >= nNodes) return;                    // uniform per-wave guard (EXEC stays all-1s)

  const float* xrow = x + (size_t)(mBase + l15) * IN_CH + 2 * hi;

  v8f c0 = {}, c1 = {}, c2 = {}, c3 = {};

#pragma unroll 4
  for (int kk = 0; kk < IN_CH; kk += 4) {
    float2 a2 = *(const float2*)(xrow + kk);
    v2f a; a[0] = a2.x; a[1] = a2.y;

    const float2* bp = &sW[((kk >> 1) + hi) * HID + l15];
    float2 w0 = bp[0], w1 = bp[16], w2 = bp[32], w3 = bp[48];

    v2f b;
    b[0] = w0.x; b[1] = w0.y;
    c0 = __builtin_amdgcn_wmma_f32_16x16x4_f32(false, a, false, b, (short)0, c0, false, false);
    b[0] = w1.x; b[1] = w1.y;
    c1 = __builtin_amdgcn_wmma_f32_16x16x4_f32(false, a, false, b, (short)0, c1, false, false);
    b[0] = w2.x; b[1] = w2.y;
    c2 = __builtin_amdgcn_wmma_f32_16x16x4_f32(false, a, false, b, (short)0, c2, false, false);
    b[0] = w3.x; b[1] = w3.y;
    c3 = __builtin_amdgcn_wmma_f32_16x16x4_f32(false, a, false, b, (short)0, c3, false, false);
  }

#pragma unroll
  for (int v = 0; v < 8; ++v) {
    float* o = xw + (size_t)(mBase + v + 8 * hi) * HID + l15;
    o[0]  = c0[v];
    o[16] = c1[v];
    o[32] = c2[v];
    o[48] = c3[v];
  }
}

// ---------------------------------------------------------------------------
// 3) scatter-add: agg[dst] += xw[src]; 64 threads per edge, one f32 atomic
//    per channel. xw and agg are L2-resident (12.8 MB each vs 192 MB L2).
// ---------------------------------------------------------------------------
__global__ void __launch_bounds__(256) scatter_add(const float* __restrict__ xw,
                                                   const long long* __restrict__ ei,
                                                   float* __restrict__ agg,
                                                   int nEdges) {
  int t = blockIdx.x * blockDim.x + threadIdx.x;
  int e = t >> 6;
  int c = t & (HID - 1);
  if (e >= nEdges) return;
  int s = (int)ei[e];            // edge_index[0][e]
  int d = (int)ei[nEdges + e];   // edge_index[1][e]
  const float* srow = xw + (size_t)s * HID;
  if (c == 0) __builtin_prefetch(srow, 0, 0);   // global_prefetch_b8
  float v = srow[c];
  atomicAdd(&agg[(size_t)d * HID + c], v);
}

// ---------------------------------------------------------------------------
// 4) out = PReLU(agg + bias), in place, float4 vectorized
// ---------------------------------------------------------------------------
__global__ void __launch_bounds__(256) bias_prelu(float* __restrict__ out,
                                                  const float* __restrict__ bias,
                                                  const float* __restrict__ pa,
                                                  int n4) {
  int i = blockIdx.x * blockDim.x + threadIdx.x;
  if (i >= n4) return;
  int c = (i * 4) & (HID - 1);
  float4 v = ((float4*)out)[i];
  float4 b = *(const float4*)(bias + c);
  float4 a = *(const float4*)(pa + c);
  v.x += b.x; v.y += b.y; v.z += b.z; v.w += b.w;
  v.x = v.x > 0.f ? v.x : a.x * v.x;
  v.y = v.y > 0.f ? v.y : a.y * v.y;
  v.z = v.z > 0.f ? v.z : a.z * v.z;
  v.w = v.w > 0.f ? v.w : a.w * v.w;
  ((float4*)out)[i] = v;
}

// ---------------------------------------------------------------------------
extern "C" void kernel_launch(void* const* d_in, const int* in_sizes, int n_in,
                              void* d_out, int out_size, void* d_ws, size_t ws_size,
                              hipStream_t stream) {
  const float*     x    = (const float*)d_in[0];
  const long long* ei   = (const long long*)d_in[1];  // int64 [2, E]
  const float*     W    = (const float*)d_in[2];
  const float*     bias = (const float*)d_in[3];
  const float*     pa   = (const float*)d_in[4];
  float*           out  = (float*)d_out;

  const int nNodes = in_sizes[0] / IN_CH;
  const int nEdges = in_sizes[1] / 2;
  float* xw = (float*)d_ws;                 // [nNodes, HID] scratch (12.8 MB)

  const int outElems = nNodes * HID;        // == out_size
  const int n4 = outElems / 4;

  // 1) zero accumulator (d_out) — harness does not re-poison between replays
  zero_f4<<<(n4 + 255) / 256, 256, 0, stream>>>((float4*)out, n4);

  // 2) GEMM: xw = x @ W (WMMA fp32)
  const int mTiles = (nNodes + 15) / 16;
  const int gBlocks = (mTiles + 7) / 8;     // 8 waves (tiles) per block
  gemm_xw_wmma<<<gBlocks, 256, 0, stream>>>(x, W, xw, nNodes);

  // 3) scatter-add over edges into d_out
  const long long work = (long long)nEdges * HID;
  const int sBlocks = (int)((work + 255) / 256);
  scatter_add<<<sBlocks, 256, 0, stream>>>(xw, ei, out, nEdges);

  // 4) bias + PReLU in place
  bias_prelu<<<(n4 + 255) / 256, 256, 0, stream>>>(out, bias, pa, n4);
}